// OpticalFlowLoss_31233002176844
// MI455X (gfx1250) — compile-verified
//
#include <hip/hip_runtime.h>

typedef __attribute__((ext_vector_type(2))) float v2f;
typedef __attribute__((ext_vector_type(8))) float v8f;

namespace {
constexpr int B_ = 8, T_ = 8, H_ = 256, W_ = 256;
constexpr int HW    = H_ * W_;
constexpr int FRAME = 3 * HW;             // floats per frame
constexpr int PAIRS = B_ * (T_ - 1);      // 56
constexpr int TILE_W = 64, TILE_H = 16;
constexpr int TX  = W_ / TILE_W;          // 4
constexpr int TYC = H_ / TILE_H;          // 16
constexpr int HALO_W = TILE_W + 2;        // 66
constexpr int HALO_H = TILE_H + 2;        // 18
constexpr int NT = 256;                   // 8 wave32 per block
constexpr int NBLOCKS = PAIRS * TX * TYC; // 3584
}

__device__ __forceinline__ float gray3(float r, float g, float b) {
    return fmaf(0.2989f, r, fmaf(0.587f, g, 0.114f * b));
}

__global__ __launch_bounds__(NT)
void flow_loss_main(const float* __restrict__ pred, const float* __restrict__ gt,
                    float* __restrict__ partial) {
    __shared__ float sP[HALO_H * HALO_W];
    __shared__ float sG[HALO_H * HALO_W];
    __shared__ float sWave[NT / 32];

    const int blk  = blockIdx.x;
    const int tile = blk & (TX * TYC - 1);
    const int pair = blk / (TX * TYC);
    const int tx = tile & (TX - 1);
    const int ty = tile / TX;
    const int b  = pair / (T_ - 1);
    const int t  = pair % (T_ - 1);

    const float* __restrict__ p1 = pred + (size_t)(b * T_ + t) * FRAME;
    const float* __restrict__ p2 = p1 + FRAME;
    const float* __restrict__ g1 = gt   + (size_t)(b * T_ + t) * FRAME;
    const float* __restrict__ g2 = g1 + FRAME;

    const int x0 = tx * TILE_W, y0 = ty * TILE_H;
    const int tid = threadIdx.x;

    // Prefetch frame t+1 center lines (overlaps with the LDS halo phase).
    {
        const int o = (y0 + (tid >> 6)) * W_ + x0 + (tid & 63);
        __builtin_prefetch(p2 + o, 0, 0);
        __builtin_prefetch(g2 + o, 0, 0);
    }

    // Stage gray(frame t) with 1-px zero-padded halo into LDS (pred & gt).
    for (int i = tid; i < HALO_H * HALO_W; i += NT) {
        const int hy = i / HALO_W, hx = i - hy * HALO_W;
        const int gy = y0 + hy - 1, gx = x0 + hx - 1;
        float gp = 0.f, gg = 0.f;
        if ((unsigned)gy < (unsigned)H_ && (unsigned)gx < (unsigned)W_) {
            const int o = gy * W_ + gx;
            gp = gray3(p1[o], p1[o + HW], p1[o + 2 * HW]);
            gg = gray3(g1[o], g1[o + HW], g1[o + 2 * HW]);
        }
        sP[i] = gp;
        sG[i] = gg;
    }
    __syncthreads();

    float acc = 0.f;
    const int lx = tid & (TILE_W - 1);
    for (int ly = tid / TILE_W; ly < TILE_H; ly += NT / TILE_W) {
        const int o = (y0 + ly) * W_ + (x0 + lx);

        const float gp2  = gray3(p2[o], p2[o + HW], p2[o + 2 * HW]);
        const float g2r = g2[o], g2g = g2[o + HW], g2b = g2[o + 2 * HW];
        const float gg2  = gray3(g2r, g2g, g2b);
        const float g1r = g1[o], g1g = g1[o + HW], g1b = g1[o + 2 * HW];
        const float mm = (fabsf(g2r - g1r) + fabsf(g2g - g1g) + fabsf(g2b - g1b)) * (1.f / 3.f);

        const int s = (ly + 1) * HALO_W + (lx + 1);

        // pred: Sobel + flow
        float a00 = sP[s - HALO_W - 1], a01 = sP[s - HALO_W], a02 = sP[s - HALO_W + 1];
        float a10 = sP[s - 1],          a11 = sP[s],          a12 = sP[s + 1];
        float a20 = sP[s + HALO_W - 1], a21 = sP[s + HALO_W], a22 = sP[s + HALO_W + 1];
        float IxP = (a00 - a02) + 2.f * (a10 - a12) + (a20 - a22);
        float IyP = (a00 + 2.f * a01 + a02) - (a20 + 2.f * a21 + a22);
        float ItP = gp2 - a11;
        float fP  = -ItP / (IxP * IxP + IyP * IyP + 1e-3f);
        float uP = IxP * fP, vP = IyP * fP;

        // gt: Sobel + flow
        float b00 = sG[s - HALO_W - 1], b01 = sG[s - HALO_W], b02 = sG[s - HALO_W + 1];
        float b10 = sG[s - 1],          b11 = sG[s],          b12 = sG[s + 1];
        float b20 = sG[s + HALO_W - 1], b21 = sG[s + HALO_W], b22 = sG[s + HALO_W + 1];
        float IxG = (b00 - b02) + 2.f * (b10 - b12) + (b20 - b22);
        float IyG = (b00 + 2.f * b01 + b02) - (b20 + 2.f * b21 + b22);
        float ItG = gg2 - b11;
        float fG  = -ItG / (IxG * IxG + IyG * IyG + 1e-3f);
        float uG = IxG * fG, vG = IyG * fG;

        acc += (fabsf(uP - uG) + fabsf(vP - vG)) * mm;
    }

    // wave32 reduction, then cross-wave via LDS; one slot per block (deterministic).
    #pragma unroll
    for (int off = 16; off > 0; off >>= 1) acc += __shfl_down(acc, off, 32);
    if ((tid & 31) == 0) sWave[tid >> 5] = acc;
    __syncthreads();
    if (tid == 0) {
        float s = 0.f;
        #pragma unroll
        for (int i = 0; i < NT / 32; ++i) s += sWave[i];
        partial[blk] = s;
    }
}

// Final 3584 -> 1 reduction through the matrix unit:
// A (16x4 f32, per-lane v2f accumulators) x B (4x16 all-ones) + 0.
// D[m][n] = rowsum(m); per-lane sum of the 8 D VGPRs gives rows 0-7
// (lanes 0-15) or rows 8-15 (lanes 16-31); one shfl_xor(16) closes it.
__global__ __launch_bounds__(32)
void flow_loss_reduce(const float* __restrict__ partial, int n,
                      float* __restrict__ out, float scale) {
    const int lane = threadIdx.x;
    float a0 = 0.f, a1 = 0.f;
    for (int i = 2 * lane; i < n; i += 64) {
        a0 += partial[i];
        if (i + 1 < n) a1 += partial[i + 1];
    }
    v2f A; A.x = a0;  A.y = a1;
    v2f Bm; Bm.x = 1.f; Bm.y = 1.f;   // all-ones B: layout-invariant
    v8f C = {};
    v8f D = __builtin_amdgcn_wmma_f32_16x16x4_f32(
        /*neg_a=*/false, A, /*neg_b=*/false, Bm,
        /*c_mod=*/(short)0, C, /*reuse_a=*/false, /*reuse_b=*/false);
    float s = D[0] + D[1] + D[2] + D[3] + D[4] + D[5] + D[6] + D[7];
    s += __shfl_xor(s, 16, 32);
    if (lane == 0) out[0] = s * scale;
}

extern "C" void kernel_launch(void* const* d_in, const int* in_sizes, int n_in,
                              void* d_out, int out_size, void* d_ws, size_t ws_size,
                              hipStream_t stream) {
    (void)in_sizes; (void)n_in; (void)out_size; (void)ws_size;
    const float* pred = (const float*)d_in[0];
    const float* gt   = (const float*)d_in[1];
    float* out        = (float*)d_out;
    float* partial    = (float*)d_ws;   // NBLOCKS * 4 bytes = 14 KB

    hipLaunchKernelGGL(flow_loss_main, dim3(NBLOCKS), dim3(NT), 0, stream,
                       pred, gt, partial);

    const float scale = 1.0f / (float)((size_t)PAIRS * 2 * H_ * W_); // mean over [B,T-1,2,H,W]
    hipLaunchKernelGGL(flow_loss_reduce, dim3(1), dim3(32), 0, stream,
                       partial, NBLOCKS, out, scale);
}